// DeformConv2d_51178830299148
// MI455X (gfx1250) — compile-verified
//
#include <hip/hip_runtime.h>
#include <hip/hip_bf16.h>
#include <math.h>

typedef __attribute__((ext_vector_type(2))) float v2f;
typedef __attribute__((ext_vector_type(8))) float v8f;

#define B_    4
#define C_    64
#define H_    128
#define W_    128
#define HW_   (H_*W_)
#define NPT   9            // 3x3 samples
#define KTOT  576          // C_*9
#define CBLK  16
#define KLOC  (CBLK*NPT)   // 144 K-values per channel block (72 pairs)

// LDS layout (in floats, all offsets 8B-aligned):
//  bt2   : KLOC*W_        = 18432   B tile, pair layout [72][128][2]
//  wt2   : KLOC*64        = 9216    A tile, pair layout [72][rows][2]
//  pm    : 128*28         = 3584    offset/mask channels per pixel
//  metaW : 128*9*4        = 4608    bilinear weights * modulation (0 if corner invalid)
//  metaS : 128*9*4 u16    = 2304    clamped linear offsets into channel plane
#define OFF_WT    (KLOC*W_)
#define OFF_PM    (OFF_WT + KLOC*64)
#define OFF_MW    (OFF_PM + 128*28)
#define OFF_MS    (OFF_MW + 128*NPT*4)
#define SMEM_FLT  (OFF_MS + 128*NPT*2)   // 38144 floats = 152576 B

__global__ __launch_bounds__(256) void deform_fused_kernel(
    const float* __restrict__ x,
    const float* __restrict__ p_w, const float* __restrict__ p_b,
    const float* __restrict__ m_w, const float* __restrict__ m_b,
    const float* __restrict__ conv_w,
    float* __restrict__ out)
{
  extern __shared__ float smem[];
  float*        bts   = smem;                       // scalar view of B tile
  v2f*          bt2   = (v2f*)smem;                 // pair view
  float*        wts   = smem + OFF_WT;              // scalar view of A tile
  v2f*          wt2   = (v2f*)(smem + OFF_WT);      // pair view
  float*        pm    = smem + OFF_PM;
  float*        metaW = smem + OFF_MW;
  unsigned int* metaS = (unsigned int*)(smem + OFF_MS);   // 2 u16 per uint

  const int tid  = threadIdx.x;
  const int lane = tid & 31;
  const int wave = tid >> 5;
  const int half = lane >> 4;     // K-half of the fragment (0/1)
  const int lr   = lane & 15;     // row (A) / column (B,D) within tile
  const int bh   = blockIdx.x;
  const int b    = bh / H_;
  const int h    = bh % H_;
  const int n0   = wave * 16;     // this wave's 16-pixel tile base

  const float* xb = x + (size_t)b * C_ * HW_;

  // ---------------- Stage 1: offset(18ch)+mask(9ch) 3x3 conv via WMMA -------
  v8f accP0 = {};   // channels 0..15  (p_w rows)
  v8f accP1 = {};   // channels 16..31 (16,17 p_w ; 18..26 m_w ; rest zero)
  for (int cb = 0; cb < C_ / CBLK; ++cb) {
    const int c0 = cb * CBLK;
    // A tile: rows 0..31 packed from p_w / m_w / zeros, pair layout [72][32][2]
    for (int e = tid; e < KLOC * 32; e += 256) {
      int kl = e % KLOC;
      int r  = e / KLOC;
      int kg = c0 * 9 + kl;
      float v = 0.f;
      if (r < 18)      v = p_w[(size_t)r * KTOT + kg];
      else if (r < 27) v = m_w[(size_t)(r - 18) * KTOT + kg];
      wts[((kl >> 1) * 32 + r) * 2 + (kl & 1)] = v;
    }
    // B tile: zero-padded im2col; one (channel, pixel) entry -> 9 taps
    for (int e = tid; e < CBLK * W_; e += 256) {
      int cl  = e >> 7;            // channel within block
      int pix = e & 127;
      const float* xp0 = xb + (size_t)(c0 + cl) * HW_;
      int cm1 = (pix > 0)   ? pix - 1 : 0;     // clamped columns (safe addrs)
      int cp1 = (pix < 127) ? pix + 1 : 127;
      bool okm = (pix > 0), okp = (pix < 127);
#pragma unroll
      for (int rr = 0; rr < 3; ++rr) {
        int gr = h + rr - 1;
        bool rowok = (gr >= 0) && (gr < H_);
        const float* xr = xp0 + (size_t)(rowok ? gr : 0) * W_;
        float vm1 = (rowok && okm) ? 1.f : 0.f;  // fold validity via select below
        float t0 = xr[cm1], t1 = xr[pix], t2 = xr[cp1];
        float v0 = (rowok && okm) ? t0 : 0.f;
        float v1 = rowok          ? t1 : 0.f;
        float v2 = (rowok && okp) ? t2 : 0.f;
        (void)vm1;
        int klb = cl * 9 + rr * 3;
        bts[(((klb + 0) >> 1) * W_ + pix) * 2 + ((klb + 0) & 1)] = v0;
        bts[(((klb + 1) >> 1) * W_ + pix) * 2 + ((klb + 1) & 1)] = v1;
        bts[(((klb + 2) >> 1) * W_ + pix) * 2 + ((klb + 2) & 1)] = v2;
      }
    }
    __syncthreads();
    {
      int bIdx = half * W_ + n0 + lr;   // pair index: [k2=half][pixel]
      int aIdx = half * 32 + lr;        // pair index: [k2=half][row]
#pragma unroll 4
      for (int kk = 0; kk < KLOC; kk += 4) {
        v2f bf = bt2[bIdx];
        v2f a0 = wt2[aIdx];
        v2f a1 = wt2[aIdx + 16];
        accP0 = __builtin_amdgcn_wmma_f32_16x16x4_f32(false, a0, false, bf, (short)0, accP0, false, false);
        accP1 = __builtin_amdgcn_wmma_f32_16x16x4_f32(false, a1, false, bf, (short)0, accP1, false, false);
        bIdx += 2 * W_;
        aIdx += 2 * 32;
      }
    }
    __syncthreads();
  }
  // bias (+sigmoid for mask) and stash per-pixel channels in LDS
  {
    int pix = n0 + lr;
#pragma unroll
    for (int r = 0; r < 8; ++r) {
      int ch0 = r + 8 * half;                 // 0..15 -> always offset channel
      pm[pix * 28 + ch0] = accP0[r] + p_b[ch0];
      int ch1 = 16 + r + 8 * half;            // 16..31
      float w = accP1[r];
      if (ch1 < 18) {
        pm[pix * 28 + ch1] = w + p_b[ch1];
      } else if (ch1 < 27) {
        float s = w + m_b[ch1 - 18];
        pm[pix * 28 + ch1] = __fdividef(1.f, 1.f + __expf(-s));   // sigmoid
      }
    }
  }
  __syncthreads();

  // ---------------- Stage 2: bilinear sampling metadata ---------------------
  // For each (pixel, n): 4 corner weights (pre-multiplied by modulation and
  // zeroed when the corner lies in the zero-pad ring / outside) and 4 clamped
  // linear offsets into a channel plane (u16).
  for (int e = tid; e < W_ * NPT; e += 256) {
    int pix = e / NPT, n = e % NPT;
    float offx = pm[pix * 28 + n];
    float offy = pm[pix * 28 + 9 + n];
    float mk   = pm[pix * 28 + 18 + n];
    float px = offx + (float)(n / 3 - 1) + (float)(h + 1);   // padded-row coord
    float py = offy + (float)(n % 3 - 1) + (float)(pix + 1); // padded-col coord
    float fx = floorf(px), fy = floorf(py);
    float fx0 = fminf(fmaxf(fx,       0.f), 129.f);
    float fx1 = fminf(fmaxf(fx + 1.f, 0.f), 129.f);
    float fy0 = fminf(fmaxf(fy,       0.f), 129.f);
    float fy1 = fminf(fmaxf(fy + 1.f, 0.f), 129.f);
    float pxc = fminf(fmaxf(px, 0.f), 129.f);
    float pyc = fminf(fmaxf(py, 0.f), 129.f);
    float glt = (1.f + (fx0 - pxc)) * (1.f + (fy0 - pyc));
    float grb = (1.f - (fx1 - pxc)) * (1.f - (fy1 - pyc));
    float glb = (1.f + (fx0 - pxc)) * (1.f - (fy1 - pyc));
    float grt = (1.f - (fx1 - pxc)) * (1.f + (fy0 - pyc));
    int x0i = (int)fx0, x1i = (int)fx1, y0i = (int)fy0, y1i = (int)fy1;
    bool vx0 = (x0i >= 1) && (x0i <= 128), vx1 = (x1i >= 1) && (x1i <= 128);
    bool vy0 = (y0i >= 1) && (y0i <= 128), vy1 = (y1i >= 1) && (y1i <= 128);
    unsigned int olt = (vx0 && vy0) ? (unsigned)((x0i - 1) * W_ + (y0i - 1)) : 0u;
    unsigned int orb = (vx1 && vy1) ? (unsigned)((x1i - 1) * W_ + (y1i - 1)) : 0u;
    unsigned int olb = (vx0 && vy1) ? (unsigned)((x0i - 1) * W_ + (y1i - 1)) : 0u;
    unsigned int ort = (vx1 && vy0) ? (unsigned)((x1i - 1) * W_ + (y0i - 1)) : 0u;
    metaW[e * 4 + 0] = (vx0 && vy0) ? glt * mk : 0.f;
    metaW[e * 4 + 1] = (vx1 && vy1) ? grb * mk : 0.f;
    metaW[e * 4 + 2] = (vx0 && vy1) ? glb * mk : 0.f;
    metaW[e * 4 + 3] = (vx1 && vy0) ? grt * mk : 0.f;
    metaS[e * 2 + 0] = olt | (orb << 16);
    metaS[e * 2 + 1] = olb | (ort << 16);
  }
  __syncthreads();

  // ---------------- Stage 3: final 64-ch GEMM via WMMA ----------------------
  v8f acc[4];
  acc[0] = (v8f){}; acc[1] = (v8f){}; acc[2] = (v8f){}; acc[3] = (v8f){};
  for (int cb = 0; cb < C_ / CBLK; ++cb) {
    const int c0 = cb * CBLK;
    // A tile: conv_w channel-block, pair layout [72][64][2]
    for (int e = tid; e < KLOC * 64; e += 256) {
      int kl = e % KLOC;
      int oc = e / KLOC;
      wts[((kl >> 1) * 64 + oc) * 2 + (kl & 1)] =
          conv_w[(size_t)oc * KTOT + c0 * 9 + kl];
    }
    // B tile: modulated deformable samples. One (pixel, n) entry per thread
    // iteration; metadata amortized across the 16 channels of the block.
    for (int e = tid; e < W_ * NPT; e += 256) {
      int pix = e / NPT, n = e % NPT;
      float w0 = metaW[e * 4 + 0], w1 = metaW[e * 4 + 1];
      float w2 = metaW[e * 4 + 2], w3 = metaW[e * 4 + 3];
      unsigned int s01 = metaS[e * 2 + 0];
      unsigned int s23 = metaS[e * 2 + 1];
      int o0 = (int)(s01 & 0xffffu), o1 = (int)(s01 >> 16);
      int o2 = (int)(s23 & 0xffffu), o3 = (int)(s23 >> 16);
      const float* xc = xb + (size_t)c0 * HW_;
#pragma unroll 4
      for (int cl = 0; cl < CBLK; ++cl) {
        float v = w0 * xc[o0] + w1 * xc[o1] + w2 * xc[o2] + w3 * xc[o3];
        int kl = cl * 9 + n;
        bts[((kl >> 1) * W_ + pix) * 2 + (kl & 1)] = v;
        xc += HW_;
      }
    }
    __syncthreads();
    {
      int bIdx = half * W_ + n0 + lr;   // pair index: [k2=half][pixel]
      int aIdx = half * 64 + lr;        // pair index: [k2=half][oc]
#pragma unroll 4
      for (int kk = 0; kk < KLOC; kk += 4) {
        v2f bf = bt2[bIdx];
        v2f a0 = wt2[aIdx];
        v2f a1 = wt2[aIdx + 16];
        v2f a2 = wt2[aIdx + 32];
        v2f a3 = wt2[aIdx + 48];
        acc[0] = __builtin_amdgcn_wmma_f32_16x16x4_f32(false, a0, false, bf, (short)0, acc[0], false, false);
        acc[1] = __builtin_amdgcn_wmma_f32_16x16x4_f32(false, a1, false, bf, (short)0, acc[1], false, false);
        acc[2] = __builtin_amdgcn_wmma_f32_16x16x4_f32(false, a2, false, bf, (short)0, acc[2], false, false);
        acc[3] = __builtin_amdgcn_wmma_f32_16x16x4_f32(false, a3, false, bf, (short)0, acc[3], false, false);
        bIdx += 2 * W_;
        aIdx += 2 * 64;
      }
    }
    __syncthreads();
  }
  // write output (B, 64, H, W)
  {
    int w = n0 + lr;
#pragma unroll
    for (int mt = 0; mt < 4; ++mt) {
#pragma unroll
      for (int r = 0; r < 8; ++r) {
        int oc = mt * 16 + r + 8 * half;
        out[(((size_t)b * C_ + oc) * H_ + h) * W_ + w] = acc[mt][r];
      }
    }
  }
}

extern "C" void kernel_launch(void* const* d_in, const int* in_sizes, int n_in,
                              void* d_out, int out_size, void* d_ws, size_t ws_size,
                              hipStream_t stream) {
  const float* x      = (const float*)d_in[0];
  const float* p_w    = (const float*)d_in[1];
  const float* p_b    = (const float*)d_in[2];
  const float* m_w    = (const float*)d_in[3];
  const float* m_b    = (const float*)d_in[4];
  const float* conv_w = (const float*)d_in[5];
  float* out = (float*)d_out;

  dim3 grid(B_ * H_);          // one workgroup per (batch, output row)
  dim3 block(256);             // 8 wave32 -> 8 x 16-pixel N-tiles = full row
  size_t shmem = (size_t)SMEM_FLT * sizeof(float);   // ~152.6 KB dynamic LDS
  hipLaunchKernelGGL(deform_fused_kernel, grid, block, shmem, stream,
                     x, p_w, p_b, m_w, m_b, conv_w, out);
}